// GENN_6468220748548
// MI455X (gfx1250) — compile-verified
//
#include <hip/hip_runtime.h>
#include <hip/hip_bf16.h>
#include <math.h>

// ---------------- problem constants (match reference) ----------------
#define NB    256          // graphs per side
#define NPG   128          // nodes per graph
#define NN    (NB * NPG)   // 32768 nodes per side
#define NE    (NN * 16)    // 524288 edges per side
#define F_IN  64
#define F1    128
#define F2    64
#define F3    32
#define TN    16
#define BINS  16

typedef __attribute__((ext_vector_type(2))) float v2f;
typedef __attribute__((ext_vector_type(8))) float v8f;

__device__ __forceinline__ float sigmoidf(float x) { return 1.0f / (1.0f + expf(-x)); }

// ---------------- utility: zero a float buffer ----------------
__global__ void zero_f32(float* __restrict__ p, long n) {
    long i = (long)blockIdx.x * blockDim.x + threadIdx.x;
    if (i < n) p[i] = 0.0f;
}

// ---------------- degree / normalization ----------------
__global__ void deg_count(const int* __restrict__ dst, float* __restrict__ deg, int e_cnt) {
    int e = blockIdx.x * blockDim.x + threadIdx.x;
    if (e < e_cnt) atomicAdd(&deg[dst[e]], 1.0f);
}

__global__ void deg_finalize(float* __restrict__ deg, int n) {
    int i = blockIdx.x * blockDim.x + threadIdx.x;
    if (i < n) deg[i] = rsqrtf(deg[i] + 1.0f);   // +1 self loop; deg >= 1 so no clamp needed
}

// ---------------- fp32 WMMA GEMM: Y[N,Fout] = X[N,K] @ W[K,Fout] ----------------
// 256 threads = 8 waves per block.  The K x 16 column tile of W is staged once in
// LDS and shared by all 8 waves; each wave owns one 16x16 output tile (128 rows /
// block).  K-loop of V_WMMA_F32_16X16X4_F32 with B-fragments from ds_load.
// A 16x4 f32 layout: lanes 0-15 hold M=lid; VGPR0 = K=(k+2*half), VGPR1 = K+1.
__global__ __launch_bounds__(256) void gemm_wmma(const float* __restrict__ X,
                                                 const float* __restrict__ W,
                                                 float* __restrict__ Y,
                                                 int K, int Fout) {
    __shared__ __align__(16) float sW[128 * 16];   // K <= 128
    int tn  = blockIdx.y * 16;
    int tid = threadIdx.x;

    // cooperative stage of W[K x 16] tile: (k, j) -> sW[k*16 + j]
    for (int i = tid; i < K * 16; i += 256) {
        int k = i >> 4, j = i & 15;
        sW[i] = W[(size_t)k * Fout + tn + j];
    }
    __syncthreads();

    int wave = tid >> 5;
    int l    = tid & 31;
    int half = l >> 4;
    int lid  = l & 15;
    int tm   = blockIdx.x * 128 + wave * 16;

    v8f c = {};
    const float* xrow = X + (size_t)(tm + lid) * K;
    for (int k = 0; k < K; k += 4) {
        int ka = k + 2 * half;
        v2f a, b;
        a.x = xrow[ka];
        a.y = xrow[ka + 1];
        b.x = sW[ka * 16 + lid];
        b.y = sW[(ka + 1) * 16 + lid];
        c = __builtin_amdgcn_wmma_f32_16x16x4_f32(false, a, false, b, (short)0, c, false, false);
    }
    for (int v = 0; v < 8; ++v)
        Y[(size_t)(tm + v + 8 * half) * Fout + tn + lid] = c[v];
}

// ---------------- edge scatter: out[d] += dinv[s]*dinv[d] * t[s] ----------------
__global__ void scatter_edges(const int* __restrict__ src, const int* __restrict__ dst,
                              const float* __restrict__ dinv, const float* __restrict__ t,
                              float* __restrict__ out, int e_cnt, int F) {
    int idx    = blockIdx.x * blockDim.x + threadIdx.x;
    int chunks = F >> 2;
    int e  = idx / chunks;
    int f4 = (idx - e * chunks) << 2;
    if (e >= e_cnt) return;
    int s = src[e], d = dst[e];
    float norm = dinv[s] * dinv[d];
    const float4 v = *reinterpret_cast<const float4*>(t + (size_t)s * F + f4);
    float* o = out + (size_t)d * F + f4;
    atomicAdd(o + 0, norm * v.x);
    atomicAdd(o + 1, norm * v.y);
    atomicAdd(o + 2, norm * v.z);
    atomicAdd(o + 3, norm * v.w);
}

// ---------------- epilogue: out = act(scat + dinv^2 * t + bias) ----------------
__global__ void gcn_epilogue(const float* __restrict__ scat, const float* __restrict__ t,
                             const float* __restrict__ dinv, const float* __restrict__ bias,
                             float* __restrict__ out, int F, int do_relu) {
    long i = (long)blockIdx.x * blockDim.x + threadIdx.x;
    if (i >= (long)NN * F) return;
    int n = (int)(i / F);
    int f = (int)(i - (long)n * F);
    float di = dinv[n];
    float v  = scat[i] + di * di * t[i] + bias[f];
    out[i] = do_relu ? fmaxf(v, 0.0f) : v;
}

// ---------------- per-graph similarity scores (WMMA from LDS) + sigmoid histogram ----
__device__ __forceinline__ v8f score_tile(const float* __restrict__ d1, const float* __restrict__ d2,
                                          int tm, int tn, int half, int lid) {
    v8f c = {};
    for (int k = 0; k < F3; k += 4) {
        int ka = k + 2 * half;
        v2f a, b;
        a.x = d1[(tm + lid) * F3 + ka];
        a.y = d1[(tm + lid) * F3 + ka + 1];
        b.x = d2[(tn + lid) * F3 + ka];     // B[k][n] = d2[n][k]  (scores = d1 @ d2^T)
        b.y = d2[(tn + lid) * F3 + ka + 1];
        c = __builtin_amdgcn_wmma_f32_16x16x4_f32(false, a, false, b, (short)0, c, false, false);
    }
    return c;
}

__global__ __launch_bounds__(256) void scores_hist(const float* __restrict__ f1,
                                                   const float* __restrict__ f2,
                                                   float* __restrict__ hist) {
    int b   = blockIdx.x;
    const float* g1 = f1 + (size_t)b * NPG * F3;   // 4096 floats (16 KB)
    const float* g2 = f2 + (size_t)b * NPG * F3;
    int tid  = threadIdx.x;
    int wave = tid >> 5;
    int l    = tid & 31;
    int half = l >> 4, lid = l & 15;

    __shared__ __align__(16) float s1[NPG * F3];
    __shared__ __align__(16) float s2[NPG * F3];
    __shared__ float red[256];
    __shared__ float lohi[2];
    __shared__ int   bins[BINS];

    // --- async-stage both 16 KB blocks into LDS (gfx1250 ASYNCcnt path) ---
    // each lane of global_load_async_to_lds_b128 moves 16 B; 1024 float4 per block,
    // 256 threads -> 4 issues/thread/matrix.
    {
        unsigned lb1 = (unsigned)(uintptr_t)(void*)s1;   // low 32 bits of a generic
        unsigned lb2 = (unsigned)(uintptr_t)(void*)s2;   // LDS pointer = LDS offset
        for (int i = tid; i < (NPG * F3) / 4; i += 256) {
            unsigned       lo1 = lb1 + (unsigned)(i * 16);
            unsigned       lo2 = lb2 + (unsigned)(i * 16);
            const float4*  a1  = reinterpret_cast<const float4*>(g1) + i;
            const float4*  a2  = reinterpret_cast<const float4*>(g2) + i;
            asm volatile("global_load_async_to_lds_b128 %0, %1, off"
                         :: "v"(lo1), "v"(a1) : "memory");
            asm volatile("global_load_async_to_lds_b128 %0, %1, off"
                         :: "v"(lo2), "v"(a2) : "memory");
        }
        asm volatile("s_wait_asynccnt 0x0" ::: "memory");
    }
    __syncthreads();

    // pass 1: raw-score min/max (sigmoid is monotone)
    float lmin =  3.4e38f, lmax = -3.4e38f;
    for (int t = wave; t < 64; t += 8) {
        int tm = (t >> 3) * 16, tn = (t & 7) * 16;
        v8f c = score_tile(s1, s2, tm, tn, half, lid);
        for (int v = 0; v < 8; ++v) {
            lmin = fminf(lmin, c[v]);
            lmax = fmaxf(lmax, c[v]);
        }
    }
    red[tid] = lmin; __syncthreads();
    for (int s = 128; s > 0; s >>= 1) { if (tid < s) red[tid] = fminf(red[tid], red[tid + s]); __syncthreads(); }
    if (tid == 0) lohi[0] = red[0];
    __syncthreads();
    red[tid] = lmax; __syncthreads();
    for (int s = 128; s > 0; s >>= 1) { if (tid < s) red[tid] = fmaxf(red[tid], red[tid + s]); __syncthreads(); }
    if (tid == 0) lohi[1] = red[0];
    if (tid < BINS) bins[tid] = 0;
    __syncthreads();

    float lo    = sigmoidf(lohi[0]);
    float hi    = sigmoidf(lohi[1]);
    float scale = (float)BINS / fmaxf(hi - lo, 1e-12f);

    // pass 2: recompute tiles from LDS, sigmoid, bin via LDS atomics
    for (int t = wave; t < 64; t += 8) {
        int tm = (t >> 3) * 16, tn = (t & 7) * 16;
        v8f c = score_tile(s1, s2, tm, tn, half, lid);
        for (int v = 0; v < 8; ++v) {
            float sv  = sigmoidf(c[v]);
            int   idx = (int)floorf((sv - lo) * scale);
            idx = idx < 0 ? 0 : (idx > BINS - 1 ? BINS - 1 : idx);
            atomicAdd(&bins[idx], 1);
        }
    }
    __syncthreads();
    if (tid < BINS)
        hist[b * BINS + tid] = (float)bins[tid] * (1.0f / (float)(NPG * NPG));
}

// ---------------- SimGNN attention pooling (one wave per graph per side) ----------------
__global__ __launch_bounds__(32) void attn_pool(const float* __restrict__ f1, const float* __restrict__ f2,
                                                const float* __restrict__ attW,
                                                float* __restrict__ e1, float* __restrict__ e2) {
    int b    = blockIdx.x;
    int side = blockIdx.y;
    const float* d = (side ? f2 : f1) + (size_t)b * NPG * F3;
    float*       e = (side ? e2 : e1) + (size_t)b * F3;
    int f = threadIdx.x;

    // mean over nodes, then @ attW, tanh  (mean(d@W) == mean(d)@W)
    float mf = 0.0f;
    for (int n = 0; n < NPG; ++n) mf += d[n * F3 + f];
    mf *= (1.0f / NPG);
    __shared__ float sm[F3];
    sm[f] = mf;
    __syncthreads();
    float ctx = 0.0f;
    for (int i = 0; i < F3; ++i) ctx += sm[i] * attW[i * F3 + f];
    ctx = tanhf(ctx);

    // att[n] = sigmoid(d[n] . ctx);  e = sum_n att[n] * d[n]
    float ef = 0.0f;
    for (int n = 0; n < NPG; ++n) {
        float dnf = d[n * F3 + f];
        float p   = dnf * ctx;
        for (int off = 16; off > 0; off >>= 1) p += __shfl_xor(p, off, 32);
        ef += sigmoidf(p) * dnf;
    }
    e[f] = ef;
}

// ---------------- tensor network + scoring MLP (one wave per graph) ----------------
__global__ __launch_bounds__(32) void final_score(const float* __restrict__ e1, const float* __restrict__ e2,
                                                  const float* __restrict__ hist,
                                                  const float* __restrict__ tW, const float* __restrict__ tblock,
                                                  const float* __restrict__ tbias,
                                                  const float* __restrict__ sW1, const float* __restrict__ sb1,
                                                  const float* __restrict__ sW2, const float* __restrict__ sb2,
                                                  float* __restrict__ out) {
    int b = blockIdx.x;
    int k = threadIdx.x;
    __shared__ float se1[F3], se2[F3];
    __shared__ float feat[TN + BINS];
    __shared__ float hh[16];
    se1[k] = e1[b * F3 + k];
    se2[k] = e2[b * F3 + k];
    __syncthreads();

    if (k < TN) {
        // ts[k] = sum_ij e1[i] * tW[i][j][k] * e2[j]     tW: (F3, F3, TN) row-major
        float ts = 0.0f;
        for (int i = 0; i < F3; ++i) {
            float a = se1[i];
            const float* row = tW + ((size_t)i * F3) * TN + k;
            for (int j = 0; j < F3; ++j) ts += a * row[j * TN] * se2[j];
        }
        // block score: tblock (TN, 2*F3), cat = [e1, e2]
        float bs = 0.0f;
        for (int m = 0; m < F3; ++m)
            bs += tblock[k * (2 * F3) + m] * se1[m] + tblock[k * (2 * F3) + F3 + m] * se2[m];
        feat[k]      = fmaxf(ts + bs + tbias[k], 0.0f);
        feat[TN + k] = hist[b * BINS + k];
    }
    __syncthreads();
    if (k < 16) {
        float acc = sb1[k];
        for (int m = 0; m < TN + BINS; ++m) acc += feat[m] * sW1[m * 16 + k];
        hh[k] = fmaxf(acc, 0.0f);
    }
    __syncthreads();
    if (k == 0) {
        float acc = sb2[0];
        for (int t = 0; t < 16; ++t) acc += hh[t] * sW2[t];
        out[b] = sigmoidf(acc);
    }
}

// ---------------- host orchestration ----------------
static inline unsigned grid1d(long n, int bs) { return (unsigned)((n + bs - 1) / bs); }

extern "C" void kernel_launch(void* const* d_in, const int* in_sizes, int n_in,
                              void* d_out, int out_size, void* d_ws, size_t ws_size,
                              hipStream_t stream) {
    (void)in_sizes; (void)n_in; (void)out_size; (void)ws_size;

    const float* x1   = (const float*)d_in[0];
    const int*   ei1  = (const int*)  d_in[1];
    const float* x2   = (const float*)d_in[3];
    const int*   ei2  = (const int*)  d_in[4];
    const float* W1   = (const float*)d_in[6];  const float* b1 = (const float*)d_in[7];
    const float* W2   = (const float*)d_in[8];  const float* b2 = (const float*)d_in[9];
    const float* W3   = (const float*)d_in[10]; const float* b3 = (const float*)d_in[11];
    const float* attW = (const float*)d_in[12];
    const float* tW   = (const float*)d_in[13];
    const float* tblk = (const float*)d_in[14];
    const float* tbia = (const float*)d_in[15];
    const float* sW1  = (const float*)d_in[16]; const float* sb1 = (const float*)d_in[17];
    const float* sW2  = (const float*)d_in[18]; const float* sb2 = (const float*)d_in[19];
    float* out = (float*)d_out;

    // workspace layout (floats)
    float* ws   = (float*)d_ws;
    float* dinv = ws;                 ws += NN;
    float* tmp  = ws;                 ws += (size_t)NN * F1;
    float* scat = ws;                 ws += (size_t)NN * F1;
    float* hbuf = ws;                 ws += (size_t)NN * F1;
    float* f1   = ws;                 ws += (size_t)NN * F3;
    float* f2   = ws;                 ws += (size_t)NN * F3;
    float* hist = ws;                 ws += NB * BINS;
    float* e1   = ws;                 ws += NB * F3;
    float* e2   = ws;                 ws += NB * F3;

    const float* xs[2]  = { x1, x2 };
    const int*   eis[2] = { ei1, ei2 };
    float*       fs[2]  = { f1, f2 };

    for (int side = 0; side < 2; ++side) {
        const float* x   = xs[side];
        const int*   src = eis[side];
        const int*   dst = eis[side] + NE;
        float*       fo  = fs[side];

        // degree -> dinv
        zero_f32    <<<grid1d(NN, 256), 256, 0, stream>>>(dinv, NN);
        deg_count   <<<grid1d(NE, 256), 256, 0, stream>>>(dst, dinv, NE);
        deg_finalize<<<grid1d(NN, 256), 256, 0, stream>>>(dinv, NN);

        // layer 1: 64 -> 128, relu
        gemm_wmma   <<<dim3(NN / 128, F1 / 16), 256, 0, stream>>>(x, W1, tmp, F_IN, F1);
        zero_f32    <<<grid1d((long)NN * F1, 256), 256, 0, stream>>>(scat, (long)NN * F1);
        scatter_edges<<<grid1d((long)NE * (F1 / 4), 256), 256, 0, stream>>>(src, dst, dinv, tmp, scat, NE, F1);
        gcn_epilogue<<<grid1d((long)NN * F1, 256), 256, 0, stream>>>(scat, tmp, dinv, b1, hbuf, F1, 1);

        // layer 2: 128 -> 64, relu
        gemm_wmma   <<<dim3(NN / 128, F2 / 16), 256, 0, stream>>>(hbuf, W2, tmp, F1, F2);
        zero_f32    <<<grid1d((long)NN * F2, 256), 256, 0, stream>>>(scat, (long)NN * F2);
        scatter_edges<<<grid1d((long)NE * (F2 / 4), 256), 256, 0, stream>>>(src, dst, dinv, tmp, scat, NE, F2);
        gcn_epilogue<<<grid1d((long)NN * F2, 256), 256, 0, stream>>>(scat, tmp, dinv, b2, hbuf, F2, 1);

        // layer 3: 64 -> 32, no relu
        gemm_wmma   <<<dim3(NN / 128, F3 / 16), 256, 0, stream>>>(hbuf, W3, tmp, F2, F3);
        zero_f32    <<<grid1d((long)NN * F3, 256), 256, 0, stream>>>(scat, (long)NN * F3);
        scatter_edges<<<grid1d((long)NE * (F3 / 4), 256), 256, 0, stream>>>(src, dst, dinv, tmp, scat, NE, F3);
        gcn_epilogue<<<grid1d((long)NN * F3, 256), 256, 0, stream>>>(scat, tmp, dinv, b3, fo, F3, 0);
    }

    // per-graph similarity histogram (async LDS staging + WMMA tiles, two passes)
    scores_hist<<<NB, 256, 0, stream>>>(f1, f2, hist);

    // attention pooling for both sides
    attn_pool<<<dim3(NB, 2), 32, 0, stream>>>(f1, f2, attW, e1, e2);

    // tensor network + scoring MLP
    final_score<<<NB, 32, 0, stream>>>(e1, e2, hist, tW, tblk, tbia, sW1, sb1, sW2, sb2, out);
}